// autoencoder_20272245637753
// MI455X (gfx1250) — compile-verified
//
#include <hip/hip_runtime.h>

// ---------------------------------------------------------------------------
// MI455X / gfx1250 implementation.
// MLP chain via fp32 WMMA GEMM (v_wmma_f32_16x16x4_f32); pdist Gram matrices
// via register-blocked WMMA with double-buffered TDM (tensor_load_to_lds)
// tile staging overlapped with compute.
// ---------------------------------------------------------------------------

typedef __attribute__((ext_vector_type(2))) float v2f;
typedef __attribute__((ext_vector_type(8))) float v8f;
typedef unsigned int u32x4 __attribute__((ext_vector_type(4)));
typedef int i32x4 __attribute__((ext_vector_type(4)));
typedef int i32x8 __attribute__((ext_vector_type(8)));

namespace {
constexpr int BATCH = 4096;
// ---- generic GEMM tiling ----
constexpr int TM = 64;        // block tile M
constexpr int TN = 64;        // block tile N
constexpr int KC = 32;        // K chunk staged in LDS
constexpr int LDP = KC + 4;   // padded LDS stride (TDM pad: 32DW interval, 4DW)
// ---- pdist tiling ----
constexpr int PTM = 64;       // rows (i)
constexpr int PTN = 128;      // cols (j)
constexpr int PKC = 16;       // K chunk (784 = 49*16, no tail)
constexpr int PLDP = PKC + 4; // padded stride = 20 floats (TDM pad: 16DW, 4DW)
constexpr size_t NPAIR = (size_t)BATCH * (BATCH - 1) / 2;
}

#if defined(__has_builtin)
#if __has_builtin(__builtin_amdgcn_tensor_load_to_lds) && \
    __has_builtin(__builtin_amdgcn_s_wait_tensorcnt)
#define USE_TDM 1
#endif
#endif
#ifndef USE_TDM
#define USE_TDM 0
#endif

__device__ __forceinline__ float act_apply(int act, float v) {
  if (act == 1) return v > 0.f ? v : 0.01f * v;   // leaky relu
  if (act == 2) return 1.f / (1.f + __expf(-v));  // sigmoid
  return v;
}

#if USE_TDM
// Build a 2D Tensor-DMA descriptor and issue TENSOR_LOAD_TO_LDS.
// Tile = tile_d1 rows x tile_d0 elements (fp32), row stride = stride elems.
// Reads beyond (tensor_d0, tensor_d1) return zero (edge/tail padding).
// pad_code: LDS padding interval code (3 = every 16 DW, 4 = every 32 DW),
// pad amount fixed at 4 DWORDs -> row stride tile_d0 + 4 floats.
__device__ __forceinline__ void tdm_load_2d(const float* gaddr, unsigned lds_off,
                                            unsigned tensor_d0, unsigned tensor_d1,
                                            unsigned tile_d0, unsigned tile_d1,
                                            unsigned long long stride,
                                            unsigned pad_code) {
  unsigned long long ga = (unsigned long long)(uintptr_t)gaddr;
  u32x4 g0;
  g0[0] = 1u;                                   // count=1, user descriptor
  g0[1] = lds_off;                              // lds_addr (bytes)
  g0[2] = (unsigned)(ga & 0xFFFFFFFFu);         // global_addr[31:0]
  g0[3] = (unsigned)((ga >> 32) & 0x01FFFFFFu)  // global_addr[56:32]
          | (2u << 30);                         // type = 2 ("image")
  i32x8 g1;
  g1[0] = (int)((2u << 16)                      // data_size = 4 bytes
                | (1u << 20)                    // pad_enable
                | (pad_code << 22)              // pad_interval
                | (3u << 25));                  // pad_amount: 4 DWORDs
  g1[1] = (int)(tensor_d0 << 16);               // [63:48] tensor_dim0 lo16
  g1[2] = (int)((tensor_d0 >> 16) | (tensor_d1 << 16));
  g1[3] = (int)((tensor_d1 >> 16) | (tile_d0 << 16));
  g1[4] = (int)(tile_d1 & 0xFFFFu);             // tile_dim1 (tile_dim2 = 0)
  g1[5] = (int)(stride & 0xFFFFFFFFull);        // tensor_dim0_stride[31:0]
  g1[6] = (int)((stride >> 32) & 0xFFFFull);    // stride[47:32], dim1_stride=0
  g1[7] = 0;
  i32x4 z4 = {0, 0, 0, 0};
#if __clang_major__ >= 23
  i32x8 z8 = {0, 0, 0, 0, 0, 0, 0, 0};
  __builtin_amdgcn_tensor_load_to_lds(g0, g1, z4, z4, z8, 0);
#else
  __builtin_amdgcn_tensor_load_to_lds(g0, g1, z4, z4, 0);
#endif
}

#define LDS_ESCAPE(p0_, p1_)                            \
  do {                                                  \
    void* e0_ = (void*)(p0_);                           \
    void* e1_ = (void*)(p1_);                           \
    asm volatile("" : "+r"(e0_), "+r"(e1_)::"memory");  \
  } while (0)
#endif  // USE_TDM

// ---------------------------------------------------------------------------
// Generic GEMM: C[M,N] = act(A[M,K] * W[N,K]^T + bias[N])
// 256 threads = 8 waves (2x4), each wave: 32x16 sub-tile (2 accumulators).
// Tiles staged by TDM (zero-fill beyond tensor dims covers N/K edges).
// ---------------------------------------------------------------------------
__global__ __launch_bounds__(256) void gemm_wmma_kernel(
    const float* __restrict__ A, const float* __restrict__ W,
    const float* __restrict__ bias, float* __restrict__ C,
    int M, int N, int K, int act) {
  __shared__ float As[TM][LDP];
  __shared__ float Bs[TN][LDP];

  const int tid  = threadIdx.x;
  const int lane = tid & 31;
  const int wave = tid >> 5;
  const int wm   = wave >> 2;
  const int wn   = wave & 3;
  const int m0   = blockIdx.y * TM;
  const int n0   = blockIdx.x * TN;

  const int lr      = lane & 15;
  const int halfsel = (lane >> 4) << 1;

  v8f acc0 = {};
  v8f acc1 = {};

#if USE_TDM
  const unsigned as_off = (unsigned)(uintptr_t)(&As[0][0]);
  const unsigned bs_off = (unsigned)(uintptr_t)(&Bs[0][0]);
#endif

  for (int k0 = 0; k0 < K; k0 += KC) {
    __syncthreads();  // WAR: previous chunk fully consumed
#if USE_TDM
    if (wave == 0)
      tdm_load_2d(A + (size_t)m0 * K + k0, as_off,
                  (unsigned)(K - k0), (unsigned)(M - m0), KC, TM,
                  (unsigned long long)K, 4u);
    if (wave == 1)
      tdm_load_2d(W + (size_t)n0 * K + k0, bs_off,
                  (unsigned)(K - k0), (unsigned)(N - n0), KC, TN,
                  (unsigned long long)K, 4u);
    if (wave < 2) __builtin_amdgcn_s_wait_tensorcnt(0);
    LDS_ESCAPE(As, Bs);
#else
    for (int t = tid; t < TM * KC; t += 256) {
      int r = t >> 5, c = t & 31;
      int gm = m0 + r, gk = k0 + c;
      As[r][c] = (gm < M && gk < K) ? A[(size_t)gm * K + gk] : 0.f;
    }
    for (int t = tid; t < TN * KC; t += 256) {
      int r = t >> 5, c = t & 31;
      int gn = n0 + r, gk = k0 + c;
      Bs[r][c] = (gn < N && gk < K) ? W[(size_t)gn * K + gk] : 0.f;
    }
#endif
    __syncthreads();

#pragma unroll
    for (int kk = 0; kk < KC; kk += 4) {
      const int kc = kk + halfsel;
      v2f a0, a1, b;
      a0[0] = As[wm * 32 + lr][kc];
      a0[1] = As[wm * 32 + lr][kc + 1];
      a1[0] = As[wm * 32 + 16 + lr][kc];
      a1[1] = As[wm * 32 + 16 + lr][kc + 1];
      b[0]  = Bs[wn * 16 + lr][kc];
      b[1]  = Bs[wn * 16 + lr][kc + 1];
      acc0 = __builtin_amdgcn_wmma_f32_16x16x4_f32(false, a0, false, b,
                                                   (short)0, acc0, false, false);
      acc1 = __builtin_amdgcn_wmma_f32_16x16x4_f32(false, a1, false, b,
                                                   (short)0, acc1, false, false);
    }
  }

  const int nn = n0 + wn * 16 + lr;
  if (nn < N) {
    const float bv = bias ? bias[nn] : 0.f;
    const int mbase0 = m0 + wm * 32 + ((lane >> 4) << 3);
    const int mbase1 = mbase0 + 16;
#pragma unroll
    for (int r = 0; r < 8; ++r) {
      int m = mbase0 + r;
      if (m < M) C[(size_t)m * N + nn] = act_apply(act, acc0[r] + bv);
      m = mbase1 + r;
      if (m < M) C[(size_t)m * N + nn] = act_apply(act, acc1[r] + bv);
    }
  }
}

// ---------------------------------------------------------------------------
// pdist epilogue helper: packed upper-triangle write
// ---------------------------------------------------------------------------
__device__ __forceinline__ void pdist_emit(const v8f& acc, int ibase, int j,
                                           float sqj, const float* __restrict__ sq,
                                           float* __restrict__ out) {
#pragma unroll
  for (int r = 0; r < 8; ++r) {
    int i = ibase + r;
    if (i < j) {
      float d2 = sq[i] + sqj - 2.f * acc[r];
      size_t idx = ((size_t)i * (2 * BATCH - i - 1)) / 2 + (size_t)(j - i - 1);
      out[idx] = sqrtf(fmaxf(d2, 0.f));
    }
  }
}

// ---------------------------------------------------------------------------
// Pairwise-distance Gram kernel. Block tile 64(i) x 128(j); 8 waves (2x4);
// each wave: 32x32 sub-tile = 2x2 WMMA fragment block (4 accumulators).
// K staged in 16-wide chunks via double-buffered TDM: chunk t+1 DMA runs
// while chunk t feeds the WMMA pipes.
// Requires: BATCH % PTN == 0, K % PKC == 0 (K = 784 = 49*16).
// ---------------------------------------------------------------------------
__global__ __launch_bounds__(256) void pdist_wmma_kernel(
    const float* __restrict__ X, const float* __restrict__ sq,
    float* __restrict__ out, int K) {
  const int bi = blockIdx.y;
  const int bj = blockIdx.x;
  const int m0 = bi * PTM;         // i block base
  const int n0 = bj * PTN;         // j block base
  if (n0 + PTN - 1 <= m0) return;  // no i<j pairs in this block

  __shared__ float As[2][PTM][PLDP];
  __shared__ float Bs[2][PTN][PLDP];

  const int tid  = threadIdx.x;
  const int lane = tid & 31;
  const int wave = tid >> 5;
  const int wm   = wave >> 2;  // 0..1  (32 rows each)
  const int wn   = wave & 3;   // 0..3  (32 cols each)

  const int lr      = lane & 15;
  const int halfsel = (lane >> 4) << 1;

  v8f acc00 = {}, acc01 = {}, acc10 = {}, acc11 = {};

  const int nchunk = K / PKC;

#if USE_TDM
  unsigned as_off[2], bs_off[2];
  as_off[0] = (unsigned)(uintptr_t)(&As[0][0][0]);
  as_off[1] = (unsigned)(uintptr_t)(&As[1][0][0]);
  bs_off[0] = (unsigned)(uintptr_t)(&Bs[0][0][0]);
  bs_off[1] = (unsigned)(uintptr_t)(&Bs[1][0][0]);
  // Prologue: stage chunk 0
  if (wave == 0)
    tdm_load_2d(X + (size_t)m0 * K, as_off[0], (unsigned)K,
                (unsigned)(BATCH - m0), PKC, PTM, (unsigned long long)K, 3u);
  if (wave == 1)
    tdm_load_2d(X + (size_t)n0 * K, bs_off[0], (unsigned)K,
                (unsigned)(BATCH - n0), PKC, PTN, (unsigned long long)K, 3u);
#endif

  for (int t = 0; t < nchunk; ++t) {
    const int cur = t & 1;
#if USE_TDM
    const int nxt = cur ^ 1;
    if (wave < 2) __builtin_amdgcn_s_wait_tensorcnt(0);  // chunk t landed
    LDS_ESCAPE(As, Bs);
    __syncthreads();  // publish chunk t; all waves done reading buffer `nxt`
    if (t + 1 < nchunk) {
      const int k1 = (t + 1) * PKC;
      if (wave == 0)
        tdm_load_2d(X + (size_t)m0 * K + k1, as_off[nxt],
                    (unsigned)(K - k1), (unsigned)(BATCH - m0), PKC, PTM,
                    (unsigned long long)K, 3u);
      if (wave == 1)
        tdm_load_2d(X + (size_t)n0 * K + k1, bs_off[nxt],
                    (unsigned)(K - k1), (unsigned)(BATCH - n0), PKC, PTN,
                    (unsigned long long)K, 3u);
    }
#else
    const int k0 = t * PKC;
    __syncthreads();  // WAR: buffer `cur` fully consumed (two iterations ago)
    for (int s = tid; s < PTM * PKC; s += 256) {
      int r = s >> 4, c = s & 15;
      As[cur][r][c] = X[(size_t)(m0 + r) * K + (k0 + c)];
    }
    for (int s = tid; s < PTN * PKC; s += 256) {
      int r = s >> 4, c = s & 15;
      Bs[cur][r][c] = X[(size_t)(n0 + r) * K + (k0 + c)];
    }
    __syncthreads();
#endif

#pragma unroll
    for (int kk = 0; kk < PKC; kk += 4) {
      const int kc = kk + halfsel;
      v2f a0, a1, b0, b1;
      a0[0] = As[cur][wm * 32 + lr][kc];
      a0[1] = As[cur][wm * 32 + lr][kc + 1];
      a1[0] = As[cur][wm * 32 + 16 + lr][kc];
      a1[1] = As[cur][wm * 32 + 16 + lr][kc + 1];
      b0[0] = Bs[cur][wn * 32 + lr][kc];
      b0[1] = Bs[cur][wn * 32 + lr][kc + 1];
      b1[0] = Bs[cur][wn * 32 + 16 + lr][kc];
      b1[1] = Bs[cur][wn * 32 + 16 + lr][kc + 1];
      acc00 = __builtin_amdgcn_wmma_f32_16x16x4_f32(false, a0, false, b0,
                                                    (short)0, acc00, false, false);
      acc01 = __builtin_amdgcn_wmma_f32_16x16x4_f32(false, a0, false, b1,
                                                    (short)0, acc01, false, false);
      acc10 = __builtin_amdgcn_wmma_f32_16x16x4_f32(false, a1, false, b0,
                                                    (short)0, acc10, false, false);
      acc11 = __builtin_amdgcn_wmma_f32_16x16x4_f32(false, a1, false, b1,
                                                    (short)0, acc11, false, false);
    }
  }

  const int jc = n0 + wn * 32 + lr;
  const int ib = m0 + wm * 32 + ((lane >> 4) << 3);
  const float sqj0 = sq[jc];
  const float sqj1 = sq[jc + 16];
  pdist_emit(acc00, ib,      jc,      sqj0, sq, out);
  pdist_emit(acc01, ib,      jc + 16, sqj1, sq, out);
  pdist_emit(acc10, ib + 16, jc,      sqj0, sq, out);
  pdist_emit(acc11, ib + 16, jc + 16, sqj1, sq, out);
}

// ---------------------------------------------------------------------------
// Per-row sum of squares (one block per row, shared reduce)
// ---------------------------------------------------------------------------
__global__ __launch_bounds__(256) void rowsq_kernel(const float* __restrict__ X,
                                                    float* __restrict__ sq,
                                                    int D) {
  __shared__ float red[256];
  const int row = blockIdx.x;
  const float* xr = X + (size_t)row * D;
  float s = 0.f;
  for (int c = threadIdx.x; c < D; c += 256) {
    float v = xr[c];
    s += v * v;
  }
  red[threadIdx.x] = s;
  __syncthreads();
#pragma unroll
  for (int off = 128; off > 0; off >>= 1) {
    if (threadIdx.x < off) red[threadIdx.x] += red[threadIdx.x + off];
    __syncthreads();
  }
  if (threadIdx.x == 0) sq[row] = red[0];
}

// ---------------------------------------------------------------------------
// Elementwise tanh
// ---------------------------------------------------------------------------
__global__ __launch_bounds__(256) void tanh_kernel(const float* __restrict__ in,
                                                   float* __restrict__ out,
                                                   int n) {
  int i = blockIdx.x * 256 + threadIdx.x;
  if (i < n) out[i] = tanhf(in[i]);
}

// ---------------------------------------------------------------------------
// Latent pdist (D == 2): one block per row i, coalesced packed writes
// ---------------------------------------------------------------------------
__global__ __launch_bounds__(256) void pdist2_kernel(const float* __restrict__ z,
                                                     float* __restrict__ out) {
  const int i = blockIdx.x;  // 0 .. BATCH-2
  const float zi0 = z[2 * i];
  const float zi1 = z[2 * i + 1];
  const float sqi = zi0 * zi0 + zi1 * zi1;
  const size_t base = ((size_t)i * (2 * BATCH - i - 1)) / 2;
  for (int j = i + 1 + threadIdx.x; j < BATCH; j += 256) {
    float zj0 = z[2 * j];
    float zj1 = z[2 * j + 1];
    float sqj = zj0 * zj0 + zj1 * zj1;
    float d2 = sqi + sqj - 2.f * (zi0 * zj0 + zi1 * zj1);
    out[base + (size_t)(j - i - 1)] = sqrtf(fmaxf(d2, 0.f));
  }
}

// ---------------------------------------------------------------------------
// Host side
// ---------------------------------------------------------------------------
extern "C" void kernel_launch(void* const* d_in, const int* in_sizes, int n_in,
                              void* d_out, int out_size, void* d_ws, size_t ws_size,
                              hipStream_t stream) {
  (void)in_sizes; (void)n_in; (void)out_size; (void)ws_size;

  const float* x       = (const float*)d_in[0];
  const float* fc1_w   = (const float*)d_in[1];
  const float* fc1_b   = (const float*)d_in[2];
  const float* fc1_1_w = (const float*)d_in[3];
  const float* fc1_1_b = (const float*)d_in[4];
  const float* fc1_2_w = (const float*)d_in[5];
  const float* fc1_2_b = (const float*)d_in[6];
  const float* fc2_w   = (const float*)d_in[7];
  const float* fc2_b   = (const float*)d_in[8];
  const float* fc3_w   = (const float*)d_in[9];
  const float* fc3_b   = (const float*)d_in[10];
  const float* fc3_1_w = (const float*)d_in[11];
  const float* fc3_1_b = (const float*)d_in[12];
  const float* fc3_2_w = (const float*)d_in[13];
  const float* fc3_2_b = (const float*)d_in[14];
  const float* fc4_w   = (const float*)d_in[15];
  const float* fc4_b   = (const float*)d_in[16];

  // Output layout (concatenated in return order)
  float* y_out    = (float*)d_out;                // [B, 784]
  float* in_diff  = y_out + (size_t)BATCH * 784;  // [NPAIR]
  float* lat_diff = in_diff + NPAIR;              // [NPAIR]
  float* out_diff = lat_diff + NPAIR;             // [NPAIR]
  float* z_out    = out_diff + NPAIR;             // [B, 2]

  // Workspace
  float* buf0 = (float*)d_ws;                     // [B, 400] max
  float* buf1 = buf0 + (size_t)BATCH * 400;       // [B, 400] max
  float* sq_x = buf1 + (size_t)BATCH * 400;       // [B]
  float* sq_y = sq_x + BATCH;                     // [B]

  const dim3 blk(256);
  auto gdim = [](int n) { return (unsigned)((n + TM - 1) / TM); };
  const unsigned gM = gdim(BATCH);  // 64

  // Encoder
  gemm_wmma_kernel<<<dim3(gdim(400), gM), blk, 0, stream>>>(x, fc1_w, fc1_b, buf0, BATCH, 400, 784, 1);
  gemm_wmma_kernel<<<dim3(gdim(200), gM), blk, 0, stream>>>(buf0, fc1_1_w, fc1_1_b, buf1, BATCH, 200, 400, 1);
  gemm_wmma_kernel<<<dim3(gdim(50),  gM), blk, 0, stream>>>(buf1, fc1_2_w, fc1_2_b, buf0, BATCH, 50, 200, 1);
  tanh_kernel<<<(BATCH * 50 + 255) / 256, blk, 0, stream>>>(buf0, buf1, BATCH * 50);
  gemm_wmma_kernel<<<dim3(gdim(2),   gM), blk, 0, stream>>>(buf1, fc2_w, fc2_b, z_out, BATCH, 2, 50, 0);

  // Decoder
  gemm_wmma_kernel<<<dim3(gdim(50),  gM), blk, 0, stream>>>(z_out, fc3_w, fc3_b, buf0, BATCH, 50, 2, 1);
  gemm_wmma_kernel<<<dim3(gdim(200), gM), blk, 0, stream>>>(buf0, fc3_1_w, fc3_1_b, buf1, BATCH, 200, 50, 1);
  gemm_wmma_kernel<<<dim3(gdim(400), gM), blk, 0, stream>>>(buf1, fc3_2_w, fc3_2_b, buf0, BATCH, 400, 200, 1);
  gemm_wmma_kernel<<<dim3(gdim(784), gM), blk, 0, stream>>>(buf0, fc4_w, fc4_b, y_out, BATCH, 784, 400, 2);

  // Row squared norms
  rowsq_kernel<<<BATCH, blk, 0, stream>>>(x, sq_x, 784);
  rowsq_kernel<<<BATCH, blk, 0, stream>>>(y_out, sq_y, 784);

  // Pairwise distances (WMMA Gram), upper-triangle block grid
  pdist_wmma_kernel<<<dim3(BATCH / PTN, BATCH / PTM), blk, 0, stream>>>(x, sq_x, in_diff, 784);
  pdist_wmma_kernel<<<dim3(BATCH / PTN, BATCH / PTM), blk, 0, stream>>>(y_out, sq_y, out_diff, 784);

  // Latent pdist (D = 2)
  pdist2_kernel<<<BATCH - 1, blk, 0, stream>>>(z_out, lat_diff);
}